// LSTMModel_11501922418886
// MI455X (gfx1250) — compile-verified
//
#include <hip/hip_runtime.h>

typedef __attribute__((ext_vector_type(2))) float v2f;
typedef __attribute__((ext_vector_type(4))) float v4f;
typedef __attribute__((ext_vector_type(8))) float v8f;

#define SEQ   64
#define BATCH 32768
#define FEAT  15
#define HID   4

#define WAVES_PER_BLOCK 8
#define ROWS_PER_WAVE   16
#define ROWS_PER_BLOCK  (WAVES_PER_BLOCK * ROWS_PER_WAVE)

static __device__ __forceinline__ v8f wmma4(v2f a, v2f b, v8f c) {
    // V_WMMA_F32_16X16X4_F32 : D(16x16 f32) = A(16x4 f32) x B(4x16 f32) + C
    return __builtin_amdgcn_wmma_f32_16x16x4_f32(
        /*neg_a=*/false, a, /*neg_b=*/false, b,
        /*c_mod=*/(short)0, c, /*reuse_a=*/false, /*reuse_b=*/false);
}

// sigmoid with hardware rcp: v_mul + v_exp_f32 + v_add + v_rcp_f32
static __device__ __forceinline__ float fast_sig(float x) {
    return __builtin_amdgcn_rcpf(1.0f + __expf(-x));
}
// tanh(x) = 2*sigmoid(2x) - 1
static __device__ __forceinline__ float fast_tanh(float x) {
    return fmaf(fast_sig(2.0f * x), 2.0f, -1.0f);
}
// relu in a single v_med3_f32
static __device__ __forceinline__ float relu1(float x) {
    return __builtin_amdgcn_fmed3f(x, 0.0f, 3.402823466e38f);
}

__global__ __launch_bounds__(WAVES_PER_BLOCK * 32)
void lstm_fused_wmma(const float* __restrict__ x,
                     const float* __restrict__ w1, const float* __restrict__ b1,
                     const float* __restrict__ w2, const float* __restrict__ b2,
                     const float* __restrict__ w_ih, const float* __restrict__ w_hh,
                     const float* __restrict__ b_ih, const float* __restrict__ b_hh,
                     const float* __restrict__ w3, const float* __restrict__ b3,
                     float* __restrict__ out)
{
    // per-wave LDS, COLUMN-major [col][row]: element (row m, col c) at c*16+m.
    // A lane's 8 C-layout values (rows 8hi..8hi+7, col n) are contiguous ->
    // two ds_store_b128; A-layout reads at col*16+row are bank-conflict-free.
    __shared__ __align__(16) float lds_s[WAVES_PER_BLOCK][ROWS_PER_WAVE * ROWS_PER_WAVE];
    __shared__ __align__(16) float lds_h[WAVES_PER_BLOCK][ROWS_PER_WAVE * ROWS_PER_WAVE];

    const int lane = threadIdx.x & 31;
    const int wave = threadIdx.x >> 5;
    const int n    = lane & 15;     // C/B-layout column
    const int hi   = lane >> 4;     // lane half: K pair (A/B) or M+8 (C)
    const int kk   = 2 * hi;        // first K index held by this lane
    const int rowBase = (blockIdx.x * WAVES_PER_BLOCK + wave) * ROWS_PER_WAVE;

    float* sbufT = lds_s[wave];
    float* hbufT = lds_h[wave];
    const int cstore = n * 16 + 8 * hi;   // column-major store base for this lane

    // ---- preload weights in B layout: VGPR0 = B[kk][n], VGPR1 = B[kk+1][n] ----
    v2f B1[4];
#pragma unroll
    for (int k = 0; k < 4; ++k) {
        const int c0 = 4 * k + kk;
        float e0 = (n < HID && c0     < FEAT) ? w1[n * FEAT + c0]     : 0.0f;
        float e1 = (n < HID && c0 + 1 < FEAT) ? w1[n * FEAT + c0 + 1] : 0.0f;
        v2f t = {e0, e1};
        B1[k] = t;
    }
    v2f B2 = { (n < HID) ? w2[n * HID + kk]     : 0.0f,
               (n < HID) ? w2[n * HID + kk + 1] : 0.0f };
    v2f B3 = { w_ih[n * HID + kk], w_ih[n * HID + kk + 1] };
    v2f B4 = { w_hh[n * HID + kk], w_hh[n * HID + kk + 1] };

    // ---- loop-invariant bias vectors, ride in as WMMA C accumulators ----
    const float bias1 = (n < HID) ? b1[n] : 0.0f;
    const float bias2 = (n < HID) ? b2[n] : 0.0f;
    const float biasg = b_ih[n] + b_hh[n];
    v8f bias1v, bias2v, biasgv;
#pragma unroll
    for (int v = 0; v < 8; ++v) { bias1v[v] = bias1; bias2v[v] = bias2; biasgv[v] = biasg; }

    // gate activation constants (per-lane, loop-invariant):
    // lanes 8..11 hold the 'g' gate -> tanh = fma(sig(2x), 2, -1); others sigmoid
    const bool  isg  = (n >= 8) && (n < 12);
    const float nascl = isg ? -2.0f : -1.0f;   // pre-negated input scale
    const float amul  = isg ?  2.0f :  1.0f;
    const float aadd  = isg ? -1.0f :  0.0f;

    // ---- init LSTM state ----
    {
        v4f z = {0.0f, 0.0f, 0.0f, 0.0f};
        *(v4f*)&hbufT[cstore]     = z;
        *(v4f*)&hbufT[cstore + 4] = z;
    }
    float creg[8];
#pragma unroll
    for (int v = 0; v < 8; ++v) creg[v] = 0.0f;

    const int r  = n;            // A-layout row owned by this lane
    const int gm = rowBase + r;  // global batch row

    // tail-feature read without OOB and without exec branches
    const int o3 = hi ? 0 : 13;

    for (int s = 0; s < SEQ; ++s) {
        const float* xrow = x + ((size_t)s * BATCH + gm) * FEAT;
        const int sp = (s + 1 < SEQ) ? s + 1 : s;    // branch-free prefetch clamp
        __builtin_prefetch(x + ((size_t)sp * BATCH + gm) * FEAT, 0, 3);

        // ---- fc1: 4 chained K=4 WMMAs, bias1 pre-loaded in accumulator ----
        v2f A1_0 = { xrow[kk],      xrow[kk + 1] };
        v2f A1_1 = { xrow[4 + kk],  xrow[4 + kk + 1] };
        v2f A1_2 = { xrow[8 + kk],  xrow[8 + kk + 1] };
        float a30 = xrow[12 + kk];
        float a31 = hi ? 0.0f : xrow[o3];
        v2f A1_3 = { a30, a31 };

        v8f acc = wmma4(A1_0, B1[0], bias1v);
        acc = wmma4(A1_1, B1[1], acc);
        acc = wmma4(A1_2, B1[2], acc);
        acc = wmma4(A1_3, B1[3], acc);

        // relu (v_med3), column-major contiguous -> 2x ds_store_b128
        {
            v4f lo = { relu1(acc[0]), relu1(acc[1]), relu1(acc[2]), relu1(acc[3]) };
            v4f hi4 = { relu1(acc[4]), relu1(acc[5]), relu1(acc[6]), relu1(acc[7]) };
            *(v4f*)&sbufT[cstore]     = lo;
            *(v4f*)&sbufT[cstore + 4] = hi4;
        }

        // ---- fc2: single K=4 WMMA, bias2 in accumulator ----
        v2f A2 = { sbufT[kk * 16 + r], sbufT[(kk + 1) * 16 + r] };
        v8f c2 = wmma4(A2, B2, bias2v);
        {
            float y[8];
#pragma unroll
            for (int v = 0; v < 8; ++v) y[v] = __expf(-c2[v]);
#pragma unroll
            for (int v = 0; v < 8; ++v) y[v] = __builtin_amdgcn_rcpf(1.0f + y[v]);
            v4f lo = { y[0], y[1], y[2], y[3] };
            v4f hi4 = { y[4], y[5], y[6], y[7] };
            *(v4f*)&sbufT[cstore]     = lo;
            *(v4f*)&sbufT[cstore + 4] = hi4;
        }

        // ---- gx = h2 @ w_ih^T + (b_ih + b_hh) ----
        v2f A3 = { sbufT[kk * 16 + r], sbufT[(kk + 1) * 16 + r] };
        v8f gacc = wmma4(A3, B3, biasgv);

        // ---- gates = gx + h @ w_hh^T (gx as accumulator) ----
        v2f A4 = { hbufT[kk * 16 + r], hbufT[(kk + 1) * 16 + r] };
        v8f gates = wmma4(A4, B4, gacc);

        // ---- activations: 8 independent exp/rcp chains (fills TRANS slots) ----
        float act[8];
#pragma unroll
        for (int v = 0; v < 8; ++v) act[v] = __expf(nascl * gates[v]);
#pragma unroll
        for (int v = 0; v < 8; ++v)
            act[v] = fmaf(__builtin_amdgcn_rcpf(1.0f + act[v]), amul, aadd);

        // ---- cell/hidden update; lane n: 0-3=i, 4-7=f, 8-11=g, 12-15=o ----
        const int base = hi * 16 + (n & 3);
        float hn[8];
#pragma unroll
        for (int v = 0; v < 8; ++v) {
            float ai = __shfl(act[v], base + 0,  32);
            float af = __shfl(act[v], base + 4,  32);
            float ag = __shfl(act[v], base + 8,  32);
            float ao = __shfl(act[v], base + 12, 32);
            float cn = fmaf(af, creg[v], ai * ag);
            creg[v]  = cn;
            hn[v] = ao * fast_tanh(cn);
        }
        {
            v4f lo = { hn[0], hn[1], hn[2], hn[3] };
            v4f hi4 = { hn[4], hn[5], hn[6], hn[7] };
            *(v4f*)&hbufT[cstore]     = lo;
            *(v4f*)&hbufT[cstore + 4] = hi4;
        }
    }

    // ---- head: out[m] = sigmoid(h_last . w3 + b3) ----
    if (hi == 0) {
        float d = b3[0];
#pragma unroll
        for (int j = 0; j < HID; ++j) d += hbufT[j * 16 + n] * w3[j];
        out[rowBase + n] = fast_sig(d);
    }
}

extern "C" void kernel_launch(void* const* d_in, const int* in_sizes, int n_in,
                              void* d_out, int out_size, void* d_ws, size_t ws_size,
                              hipStream_t stream) {
    const float* x    = (const float*)d_in[0];
    const float* w1   = (const float*)d_in[1];
    const float* b1   = (const float*)d_in[2];
    const float* w2   = (const float*)d_in[3];
    const float* b2   = (const float*)d_in[4];
    const float* w_ih = (const float*)d_in[5];
    const float* w_hh = (const float*)d_in[6];
    const float* b_ih = (const float*)d_in[7];
    const float* b_hh = (const float*)d_in[8];
    const float* w3   = (const float*)d_in[9];
    const float* b3   = (const float*)d_in[10];
    float* out = (float*)d_out;

    dim3 grid(BATCH / ROWS_PER_BLOCK);           // 256 blocks
    dim3 block(WAVES_PER_BLOCK * 32);            // 8 waves of 32
    hipLaunchKernelGGL(lstm_fused_wmma, grid, block, 0, stream,
                       x, w1, b1, w2, b2, w_ih, w_hh, b_ih, b_hh, w3, b3, out);
}